// PANTHERBase_3633542332978
// MI455X (gfx1250) — compile-verified
//
#include <hip/hip_runtime.h>
#include <hip/hip_bf16.h>
#include <math.h>

// Problem constants (from reference)
constexpr int PP = 16;      // mixture components (== WMMA N dim)
constexpr int DD = 1024;    // feature dim
constexpr int NN = 16384;   // points per batch
constexpr int BB = 4;       // batches
constexpr int LL = 3;       // EM iterations
constexpr float TAUc = 10.0f;
constexpr float EPSc = 0.1f;
constexpr float LOG2PI = 1.8378770664093453f;

typedef float v2f __attribute__((ext_vector_type(2)));
typedef float v8f __attribute__((ext_vector_type(8)));

// ---------------- workspace layout (floats) ----------------
constexpr size_t OFF_V     = 0;                       // (P,D)
constexpr size_t OFF_PI    = OFF_V     + (size_t)PP*DD;        // (B,P)
constexpr size_t OFF_MU    = OFF_PI    + (size_t)BB*PP;        // (B,P,D)
constexpr size_t OFF_SIG   = OFF_MU    + (size_t)BB*PP*DD;     // (B,P,D)
constexpr size_t OFF_INV   = OFF_SIG   + (size_t)BB*PP*DD;     // (B,P,D)  1/Sigma
constexpr size_t OFF_W     = OFF_INV   + (size_t)BB*PP*DD;     // (B,P,D)  mu/Sigma
constexpr size_t OFF_CST   = OFF_W     + (size_t)BB*PP*DD;     // (B,P)
constexpr size_t OFF_WSUM  = OFF_CST   + (size_t)BB*PP;        // (B,P)
constexpr size_t OFF_WX    = OFF_WSUM  + (size_t)BB*PP;        // (B,P,D)
constexpr size_t OFF_WXX   = OFF_WX    + (size_t)BB*PP*DD;     // (B,P,D)
constexpr size_t OFF_QQ    = OFF_WXX   + (size_t)BB*PP*DD;     // (B,N,P)

// output layout
constexpr size_t OUT_STRIDE_B = (size_t)PP + 2ull*PP*DD;       // per-batch [pi, mu, Sigma]
constexpr size_t OUT_QQ       = (size_t)BB * OUT_STRIDE_B;     // then qq (B,N,P)

// ---------------- init: V = EPS*softplus(V_), params = priors ----------------
__global__ __launch_bounds__(256)
void mog_init_kernel(const float* __restrict__ m, const float* __restrict__ V_,
                     float* __restrict__ V, float* __restrict__ pi,
                     float* __restrict__ mu, float* __restrict__ Sigma) {
    int i = blockIdx.x * blockDim.x + threadIdx.x;
    if (i < PP * DD) {
        float v_ = V_[i];
        float sp = (v_ > 0.f) ? (v_ + log1pf(expf(-v_))) : log1pf(expf(v_)); // stable softplus
        float v = EPSc * sp;
        V[i] = v;
        float mm = m[i];
        for (int b = 0; b < BB; ++b) {
            mu[(size_t)b * PP * DD + i] = mm;
            Sigma[(size_t)b * PP * DD + i] = v;
        }
    }
    if (i < BB * PP) pi[i] = 1.0f / PP;
}

// ---------------- derive: inv = 1/Sigma, W = mu*inv, cst[b,p] ----------------
__global__ __launch_bounds__(256)
void mog_derive_kernel(const float* __restrict__ pi, const float* __restrict__ mu,
                       const float* __restrict__ Sigma,
                       float* __restrict__ inv, float* __restrict__ W,
                       float* __restrict__ cst) {
    int bp = blockIdx.x;                 // b*P + p
    int tid = threadIdx.x;
    const size_t base = (size_t)bp * DD;
    float slog = 0.f, smu = 0.f;
    for (int d = tid; d < DD; d += 256) {
        float sg = Sigma[base + d];
        float iv = 1.0f / sg;
        float mv = mu[base + d];
        inv[base + d] = iv;
        W[base + d] = mv * iv;
        slog += logf(sg);
        smu  += mv * mv * iv;
    }
    __shared__ float r1[256], r2[256];
    r1[tid] = slog; r2[tid] = smu;
    __syncthreads();
    for (int s = 128; s > 0; s >>= 1) {
        if (tid < s) { r1[tid] += r1[tid + s]; r2[tid] += r2[tid + s]; }
        __syncthreads();
    }
    if (tid == 0)
        cst[bp] = -0.5f * ((float)DD * LOG2PI + r1[0] + r2[0]) + logf(pi[bp]);
}

// ---------------- E-step: WMMA f32 16x16x4 over K=D ----------------
// grid = (B, 32), block = 256 (8 waves). Each wave owns 16-row tiles.
__global__ __launch_bounds__(256)
void mog_estep_kernel(const float* __restrict__ S, const float* __restrict__ mask,
                      const float* __restrict__ Wg_all,   // (B,P,D) mu*inv
                      const float* __restrict__ Ig_all,   // (B,P,D) inv
                      const float* __restrict__ cst,      // (B,P)
                      float* __restrict__ qq,             // (B,N,P)
                      float* __restrict__ wsum) {         // (B,P)
    const int b    = blockIdx.x;
    const int lane = threadIdx.x & 31;
    const int wave = threadIdx.x >> 5;
    const int half = lane >> 4;       // 0: K 0..1, 1: K 2..3
    const int lp   = lane & 15;       // row index for A, p index for B/C

    extern __shared__ float lds_buf[];
    constexpr int LROW = DD + 4;      // pad -> conflict-free ds_load_b64
    float* Wl = lds_buf;
    float* Il = lds_buf + PP * LROW;

    // cooperative stage of B operands (P x D, L2-resident) into LDS
    const float* Wg = Wg_all + (size_t)b * PP * DD;
    const float* Ig = Ig_all + (size_t)b * PP * DD;
    for (int i = threadIdx.x * 4; i < PP * DD; i += 256 * 4) {
        int p = i >> 10, d = i & (DD - 1);
        float4 w = *(const float4*)(Wg + i);
        float4 v = *(const float4*)(Ig + i);
        *(float4*)(Wl + p * LROW + d) = w;
        *(float4*)(Il + p * LROW + d) = v;
    }
    __syncthreads();

    const float cst_l = cst[b * PP + lp];
    float wsum_acc = 0.f;

    const int tiles = NN / 16;  // 1024
    for (int t = blockIdx.y * 8 + wave; t < tiles; t += gridDim.y * 8) {
        const float* xrow = S + ((size_t)b * NN + (size_t)t * 16 + lp) * DD + half * 2;
        v8f c1 = {}; // sum_d x^2 * inv
        v8f c2 = {}; // sum_d x   * mu*inv
        #pragma unroll 4
        for (int d0 = 0; d0 < DD; d0 += 4) {
            v2f a  = *(const v2f*)(xrow + d0);
            v2f a2 = a * a;
            const int baddr = lp * LROW + d0 + half * 2;
            v2f bw = *(const v2f*)(Wl + baddr);
            v2f bi = *(const v2f*)(Il + baddr);
            c2 = __builtin_amdgcn_wmma_f32_16x16x4_f32(false, a,  false, bw, (short)0, c2, false, false);
            c1 = __builtin_amdgcn_wmma_f32_16x16x4_f32(false, a2, false, bi, (short)0, c1, false, false);
        }
        // C layout: VGPR v -> row = v + 8*half, col p = lp
        const float* mb = mask + (size_t)b * NN + (size_t)t * 16 + 8 * half;
        float* qb = qq + ((size_t)b * NN + (size_t)t * 16 + 8 * half) * PP + lp;
        #pragma unroll
        for (int v = 0; v < 8; ++v) {
            float val = cst_l + c2[v] - 0.5f * c1[v];
            // softmax over p (lanes within each 16-lane half)
            float mx = val;
            mx = fmaxf(mx, __shfl_xor(mx, 8, 16));
            mx = fmaxf(mx, __shfl_xor(mx, 4, 16));
            mx = fmaxf(mx, __shfl_xor(mx, 2, 16));
            mx = fmaxf(mx, __shfl_xor(mx, 1, 16));
            float e = __expf(val - mx);
            float sm = e;
            sm += __shfl_xor(sm, 8, 16);
            sm += __shfl_xor(sm, 4, 16);
            sm += __shfl_xor(sm, 2, 16);
            sm += __shfl_xor(sm, 1, 16);
            float q = (e / sm) * mb[v];
            qb[(size_t)v * PP] = q;
            wsum_acc += q;
        }
    }
    atomicAdd(&wsum[b * PP + lp], wsum_acc);
}

// ---------------- M-step moments: wx = qq^T S, wxx = qq^T S^2 ----------------
// grid = (B, D/128, NSPLIT), block = 256 (8 waves, wave w owns 16 cols).
constexpr int NSPLIT = 8;
__global__ __launch_bounds__(256)
void mog_mstep_kernel(const float* __restrict__ S, const float* __restrict__ qq,
                      float* __restrict__ wxsum, float* __restrict__ wxxsum) {
    const int b    = blockIdx.x;
    const int lane = threadIdx.x & 31;
    const int wave = threadIdx.x >> 5;
    const int half = lane >> 4;
    const int lp   = lane & 15;
    const int dbase = blockIdx.y * 128 + wave * 16;
    const int nstart = blockIdx.z * (NN / NSPLIT);
    const int nend   = nstart + (NN / NSPLIT);

    v8f cx = {};   // wx  accumulator: M=p, N=d
    v8f cxx = {};  // wxx accumulator
    const float* qb = qq + ((size_t)b * NN) * PP;
    const float* sb = S + (size_t)b * NN * DD;
    for (int n0 = nstart; n0 < nend; n0 += 4) {
        const int r = n0 + half * 2;
        // A = qq^T: lane lp = component p, K = rows r, r+1
        float a0 = qb[(size_t)r * PP + lp];
        float a1 = qb[(size_t)(r + 1) * PP + lp];
        v2f a = {a0, a1};
        // B = x: lane lp = column d, K = rows r, r+1
        float b0 = sb[(size_t)r * DD + dbase + lp];
        float b1 = sb[(size_t)(r + 1) * DD + dbase + lp];
        v2f bx = {b0, b1};
        v2f bxx = bx * bx;
        cx  = __builtin_amdgcn_wmma_f32_16x16x4_f32(false, a, false, bx,  (short)0, cx,  false, false);
        cxx = __builtin_amdgcn_wmma_f32_16x16x4_f32(false, a, false, bxx, (short)0, cxx, false, false);
    }
    // flush: VGPR v -> p = v + 8*half, d = dbase + lp
    #pragma unroll
    for (int v = 0; v < 8; ++v) {
        size_t idx = ((size_t)b * PP + v + 8 * half) * DD + dbase + lp;
        atomicAdd(&wxsum[idx],  cx[v]);
        atomicAdd(&wxxsum[idx], cxx[v]);
    }
}

// ---------------- finalize: MAP M-step update ----------------
__global__ __launch_bounds__(256)
void mog_finalize_kernel(const float* __restrict__ wsum, const float* __restrict__ wxsum,
                         const float* __restrict__ wxxsum,
                         const float* __restrict__ m, const float* __restrict__ V,
                         float* __restrict__ pi, float* __restrict__ mu,
                         float* __restrict__ Sigma,
                         float* __restrict__ out, int write_out) {
    const int bp = blockIdx.x;
    const int b = bp / PP, p = bp % PP;
    const int tid = threadIdx.x;
    float tot = 0.f;
    for (int j = 0; j < PP; ++j) tot += wsum[b * PP + j];
    const float wp = wsum[bp];
    const float pin = wp / tot;
    const float denom = wp + TAUc;
    const size_t base = (size_t)bp * DD;
    const size_t pbase = (size_t)p * DD;
    const size_t obase = (size_t)b * OUT_STRIDE_B;
    for (int d = tid; d < DD; d += 256) {
        float mv = m[pbase + d];
        float mun = (wxsum[base + d] + mv * TAUc) / denom;
        float sgn = (wxxsum[base + d] + (V[pbase + d] + mv * mv) * TAUc) / denom - mun * mun;
        mu[base + d] = mun;
        Sigma[base + d] = sgn;
        if (write_out) {
            out[obase + PP + pbase + d] = mun;
            out[obase + PP + (size_t)PP * DD + pbase + d] = sgn;
        }
    }
    if (tid == 0) {
        pi[bp] = pin;
        if (write_out) out[obase + p] = pin;
    }
}

// ---------------- host driver ----------------
extern "C" void kernel_launch(void* const* d_in, const int* in_sizes, int n_in,
                              void* d_out, int out_size, void* d_ws, size_t ws_size,
                              hipStream_t stream) {
    const float* S    = (const float*)d_in[0];   // (B,N,D)
    const float* mask = (const float*)d_in[1];   // (B,N)
    const float* m    = (const float*)d_in[2];   // (P,D)
    const float* V_   = (const float*)d_in[3];   // (P,D)
    float* out = (float*)d_out;
    float* ws  = (float*)d_ws;

    float* V    = ws + OFF_V;
    float* pi   = ws + OFF_PI;
    float* mu   = ws + OFF_MU;
    float* Sig  = ws + OFF_SIG;
    float* inv  = ws + OFF_INV;
    float* W    = ws + OFF_W;
    float* cst  = ws + OFF_CST;
    float* wsum = ws + OFF_WSUM;
    float* wx   = ws + OFF_WX;
    float* wxx  = ws + OFF_WXX;
    float* qqws = ws + OFF_QQ;

    // init priors
    mog_init_kernel<<<(PP * DD + 255) / 256, 256, 0, stream>>>(m, V_, V, pi, mu, Sig);

    const size_t estep_lds = 2ull * PP * (DD + 4) * sizeof(float); // ~132 KB (CDNA5: 320 KB/WG)

    for (int it = 0; it < LL; ++it) {
        mog_derive_kernel<<<BB * PP, 256, 0, stream>>>(pi, mu, Sig, inv, W, cst);

        hipMemsetAsync(wsum, 0, (size_t)BB * PP * sizeof(float), stream);
        hipMemsetAsync(wx,   0, (size_t)BB * PP * DD * sizeof(float), stream);
        hipMemsetAsync(wxx,  0, (size_t)BB * PP * DD * sizeof(float), stream);

        float* qq_it = (it == LL - 1) ? (out + OUT_QQ) : qqws;

        mog_estep_kernel<<<dim3(BB, 32), 256, estep_lds, stream>>>(
            S, mask, W, inv, cst, qq_it, wsum);

        mog_mstep_kernel<<<dim3(BB, DD / 128, NSPLIT), 256, 0, stream>>>(
            S, qq_it, wx, wxx);

        mog_finalize_kernel<<<BB * PP, 256, 0, stream>>>(
            wsum, wx, wxx, m, V, pi, mu, Sig, out, (it == LL - 1) ? 1 : 0);
    }
}